// MSD_computer_4647154614877
// MI455X (gfx1250) — compile-verified
//
#include <hip/hip_runtime.h>

typedef __attribute__((ext_vector_type(2))) float v2f;
typedef __attribute__((ext_vector_type(8))) float v8f;

#define FRAME_ELEMS 1536   // 512 * 3 floats per frame
#define TD_MAX_C    2000
#define T_STRIDE    100
#define QMAX        20     // TD_MAX / T_STRIDE

// ---------------- Kernel A: per-frame squared norms S[t] ----------------
__global__ void msd_frame_norms(const float* __restrict__ x,
                                float* __restrict__ S) {
  const int t = blockIdx.x;
  const float* f = x + (size_t)t * FRAME_ELEMS;
  float p = 0.f;
  for (int k = threadIdx.x; k < FRAME_ELEMS; k += 256) {
    float v = f[k];
    p += v * v;
  }
  __shared__ float red[256];
  red[threadIdx.x] = p;
  __syncthreads();
  for (int s = 128; s > 0; s >>= 1) {
    if ((int)threadIdx.x < s) red[threadIdx.x] += red[threadIdx.x + s];
    __syncthreads();
  }
  if (threadIdx.x == 0) S[t] = red[0];
}

// ---------------- zero the cross-term accumulator ----------------
__global__ void msd_zero(float* __restrict__ Dsum, int n) {
  int i = blockIdx.x * blockDim.x + threadIdx.x;
  if (i < n) Dsum[i] = 0.f;
}

// ---------------- Kernel B: banded cross-correlation via WMMA f32 -------
// For phase r (block.x) and origin tile cb (block.y), each wave (o =
// threadIdx.y in 0..2) computes a 16x16 f32 tile of R_r * O^T with K=1536,
// i.e. D[j,i] = dot(frame[100*j + r], frame[100*i]), rows j = 16*(cb+o)..+15,
// cols i = 16*cb..+15, then scatters the band entries q = j - i in [0,20)
// into Dsum[100*q + r] with float atomics.
__global__ void msd_band_wmma(const float* __restrict__ x,
                              float* __restrict__ Dsum,
                              int NT0, int Tframes) {
  const int r    = blockIdx.x;         // td phase 0..99
  const int cb   = blockIdx.y;         // origin (column) tile
  const int o    = threadIdx.y;        // row-tile offset 0..2
  const int lane = threadIdx.x;        // 0..31 (one wave per (r,cb,o))
  const int half = lane >> 4;
  const int ln   = lane & 15;

  const int ibase = 16 * cb;           // origin index base (columns, N)
  const int jbase = 16 * (cb + o);     // shifted-frame index base (rows, M)
  const int jmax  = Tframes / T_STRIDE - 1;  // last frame index usable (=199)

  // A-matrix 16x4 layout: lane ln holds row M=ln; halves hold K={0,1}/{2,3}.
  int jA = jbase + ln;
  jA = (jA <= jmax) ? jA : jmax;       // clamp OOB rows (masked at scatter)
  const float* pa = x + (size_t)(jA * T_STRIDE + r) * FRAME_ELEMS + 2 * half;

  // B-matrix 4x16 layout: lane ln holds column N=ln; halves hold K={0,1}/{2,3}.
  int iB = ibase + ln;
  iB = (iB <= jmax) ? iB : jmax;
  const float* pb = x + (size_t)(iB * T_STRIDE) * FRAME_ELEMS + 2 * half;

  v8f c0 = {};
  v8f c1 = {};
#pragma unroll 4
  for (int kk = 0; kk < FRAME_ELEMS; kk += 8) {
    v2f a0 = *(const v2f*)(pa + kk);
    v2f b0 = *(const v2f*)(pb + kk);
    v2f a1 = *(const v2f*)(pa + kk + 4);
    v2f b1 = *(const v2f*)(pb + kk + 4);
    // D = A(16x4,f32) x B(4x16,f32) + C ; args:
    // (neg_a, A, neg_b, B, c_mod, C, reuse_a, reuse_b)
    c0 = __builtin_amdgcn_wmma_f32_16x16x4_f32(false, a0, false, b0,
                                               (short)0, c0, false, false);
    c1 = __builtin_amdgcn_wmma_f32_16x16x4_f32(false, a1, false, b1,
                                               (short)0, c1, false, false);
  }
  v8f d = c0 + c1;

  // C/D layout: VGPR g, lanes 0-15 -> M=g, lanes 16-31 -> M=g+8; N = lane&15.
  const int i = ibase + ln;            // this lane's column (origin index)
  if (i < NT0) {
#pragma unroll
    for (int g = 0; g < 8; ++g) {
      int m = g + 8 * half;            // row inside this tile
      int j = jbase + m;               // global shifted-frame index
      int q = j - i;                   // time-delta / 100
      if (q >= 0 && q < QMAX) {
        atomicAdd(&Dsum[q * T_STRIDE + r], d[g]);
      }
    }
  }
}

// ---------------- Kernel C: combine S and cross terms ----------------
__global__ void msd_combine(const float* __restrict__ S,
                            const float* __restrict__ Dsum,
                            float* __restrict__ out, int NT0) {
  int td = blockIdx.x * blockDim.x + threadIdx.x;
  if (td >= TD_MAX_C) return;
  float s = 0.f;
  for (int i = 0; i < NT0; ++i)
    s += S[i * T_STRIDE + td] + S[i * T_STRIDE];
  const float scale = 1.0f / ((float)NT0 * (float)FRAME_ELEMS);
  out[td] = (s - 2.0f * Dsum[td]) * scale;
}

extern "C" void kernel_launch(void* const* d_in, const int* in_sizes, int n_in,
                              void* d_out, int out_size, void* d_ws, size_t ws_size,
                              hipStream_t stream) {
  const float* x = (const float*)d_in[0];
  float* out = (float*)d_out;

  const int total = in_sizes[0];                 // 20000 * 512 * 3
  const int T     = total / FRAME_ELEMS;         // 20000 frames
  const int NT0   = (T - TD_MAX_C + T_STRIDE - 1) / T_STRIDE;  // 180 origins

  // Workspace layout: S[T] then Dsum[TD_MAX]  (88 KB total)
  float* S    = (float*)d_ws;
  float* Dsum = S + T;

  // 1) per-frame squared norms
  msd_frame_norms<<<T, 256, 0, stream>>>(x, S);

  // 2) zero cross-term accumulator (ws is poisoned, and we re-run per replay)
  msd_zero<<<(TD_MAX_C + 255) / 256, 256, 0, stream>>>(Dsum, TD_MAX_C);

  // 3) banded cross-correlation: 100 phases x 12 origin tiles x 3 row tiles
  const int CB = (NT0 + 15) / 16;                // 12 column tiles
  dim3 gridB(T_STRIDE, CB, 1);
  dim3 blockB(32, 3, 1);                         // 3 wave32 per block
  msd_band_wmma<<<gridB, blockB, 0, stream>>>(x, Dsum, NT0, T);

  // 4) combine into msd[td]
  msd_combine<<<(TD_MAX_C + 255) / 256, 256, 0, stream>>>(S, Dsum, out, NT0);
}